// CausalAttention_41016937677092
// MI455X (gfx1250) — compile-verified
//
#include <hip/hip_runtime.h>
#include <math.h>
#include <stdint.h>

typedef float v2f __attribute__((ext_vector_type(2)));
typedef float v8f __attribute__((ext_vector_type(8)));

#define SEQ   2048
#define ED    1024
#define DQ    256     // collapsed per-part width (Qc / K / V)
#define NW    768     // fused QKV width
#define NBATCH 8

__device__ __forceinline__ v8f wmma_f32(v2f a, v2f b, v8f c) {
  // D = A(16x4, f32) * B(4x16, f32) + C(16x16, f32)
  return __builtin_amdgcn_wmma_f32_16x16x4_f32(
      /*neg_a=*/false, a, /*neg_b=*/false, b,
      /*c_mod=*/(short)0, c, /*reuse_a=*/false, /*reuse_b=*/false);
}

// gfx1250 async copy global->LDS (ASYNCcnt-tracked). lds_off is the byte
// offset inside the workgroup LDS allocation (low 32 bits of a generic
// pointer to __shared__); gsrc is a per-lane global address.
__device__ __forceinline__ void async_g2l_b32(uint32_t lds_off, const float* gsrc) {
  asm volatile("global_load_async_to_lds_b32 %0, %1, off"
               :: "v"(lds_off), "v"(gsrc)
               : "memory");
}
#define S_WAIT_ASYNCCNT(n) asm volatile("s_wait_asynccnt " #n ::: "memory")

// ---------------------------------------------------------------------------
// Kernel 0: build fused weight Wall(1024 x 768) = [Wq_collapsed | Wk | Wv]
// and fused bias ball(768). GQA tiling folded into the Q weight:
//   Wq_c[d, g*64+e] = sum_{j=0..3} Wq[d, (g+4j)*64 + e]
// ---------------------------------------------------------------------------
__global__ void build_wall(const float* __restrict__ Wq, const float* __restrict__ bq,
                           const float* __restrict__ Wk, const float* __restrict__ bk,
                           const float* __restrict__ Wv, const float* __restrict__ bv,
                           float* __restrict__ Wall, float* __restrict__ ball) {
  int idx = blockIdx.x * blockDim.x + threadIdx.x;
  const int NWALL = ED * NW;
  if (idx < NWALL) {
    int d = idx / NW, c = idx - d * NW;
    float v;
    if (c < 256) {
      int g = c >> 6, e = c & 63;
      v = Wq[d * ED + (g + 0) * 64 + e] + Wq[d * ED + (g + 4) * 64 + e] +
          Wq[d * ED + (g + 8) * 64 + e] + Wq[d * ED + (g + 12) * 64 + e];
    } else if (c < 512) {
      v = Wk[d * DQ + (c - 256)];
    } else {
      v = Wv[d * DQ + (c - 512)];
    }
    Wall[idx] = v;
  } else if (idx < NWALL + NW) {
    int c = idx - NWALL;
    float v;
    if (c < 256) {
      int g = c >> 6, e = c & 63;
      v = bq[(g + 0) * 64 + e] + bq[(g + 4) * 64 + e] +
          bq[(g + 8) * 64 + e] + bq[(g + 12) * 64 + e];
    } else if (c < 512) {
      v = bk[c - 256];
    } else {
      v = bv[c - 512];
    }
    ball[c] = v;
  }
}

// ---------------------------------------------------------------------------
// Kernel 1: fused QKV projection GEMM, C(16384 x 768) = X(16384x1024)*Wall + b
// 8 waves per block; wave w -> rows [m0+16w, +16), 64 cols -> 4 WMMA frags.
// B tile (KC x 64) is staged in LDS *transposed* ([n][k], stride 34 dwords:
// conflict-free scatter and conflict-free ds_load_b64 fragment reads),
// double-buffered via gfx1250 async global->LDS DMA + s_wait_asynccnt.
// ---------------------------------------------------------------------------
#define KC      32                 // K-chunk per stage
#define BT_STR  34                 // padded LDS stride (dwords)
#define BT_SZ   (64 * BT_STR)      // one buffer, dwords

__global__ __launch_bounds__(256) void qkv_gemm(const float* __restrict__ X,
                                                const float* __restrict__ Wall,
                                                const float* __restrict__ ball,
                                                float* __restrict__ Qkv) {
  __shared__ float sB[2 * BT_SZ];  // 17408 B, double-buffered B tile

  const int tid  = threadIdx.x;
  const int lane = tid & 31;
  const int wave = tid >> 5;
  const int lrow = lane & 15;
  const int lhi  = lane >> 4;

  const int n0 = blockIdx.x * 64;
  const int m0 = blockIdx.y * 128 + wave * 16;

  const uint32_t sb_base = (uint32_t)(uintptr_t)(&sB[0]);

  v8f acc[4];
  const v8f z8 = {0.f, 0.f, 0.f, 0.f, 0.f, 0.f, 0.f, 0.f};
#pragma unroll
  for (int f = 0; f < 4; ++f) acc[f] = z8;

  const float* arow = X + (size_t)(m0 + lrow) * ED;

  // --- stage helper: issue 8 async b32 copies/thread for one K-chunk -------
  // element e = tid + i*256 -> (k = e>>6, n = e&63); global read coalesced,
  // LDS write transposed to [n*BT_STR + k].
  auto stage = [&](int buf, int kc) {
#pragma unroll
    for (int i = 0; i < 8; ++i) {
      const int e = tid + i * 256;
      const int k = e >> 6;
      const int n = e & 63;
      const float* gp = Wall + (size_t)(kc + k) * NW + (n0 + n);
      const uint32_t loff =
          sb_base + (uint32_t)((buf * BT_SZ + n * BT_STR + k) * 4);
      async_g2l_b32(loff, gp);
    }
  };

  stage(0, 0);
  int cur = 0;
  for (int kc = 0; kc < ED; kc += KC) {
    if (kc + KC < ED) {
      stage(cur ^ 1, kc + KC);   // prefetch next chunk (8 more in flight)
      S_WAIT_ASYNCCNT(8);        // current chunk's 8 copies are done
    } else {
      S_WAIT_ASYNCCNT(0);
    }
    __syncthreads();             // make every wave's DMA'd data visible

    const float* Bt = &sB[cur * BT_SZ];
#pragma unroll
    for (int k = 0; k < KC; k += 4) {
      const int kk = k + 2 * lhi;                  // lane's two K indices
      v2f a = *(const v2f*)(arow + kc + kk);
#pragma unroll
      for (int f = 0; f < 4; ++f) {
        v2f bb = *(const v2f*)(&Bt[(f * 16 + lrow) * BT_STR + kk]);
        acc[f] = wmma_f32(a, bb, acc[f]);
      }
    }
    __syncthreads();             // done reading 'cur' before it is re-staged
    cur ^= 1;
  }

#pragma unroll
  for (int f = 0; f < 4; ++f) {
    const int n = n0 + f * 16 + lrow;
    const float bias = ball[n];
#pragma unroll
    for (int j = 0; j < 8; ++j) {
      const int m = m0 + j + 8 * lhi;              // C layout: vgpr j -> row
      Qkv[(size_t)m * NW + n] = acc[f][j] + bias;
    }
  }
}

// ---------------------------------------------------------------------------
// Kernel 2: flash-style causal attention for one (batch, 64-row t-tile).
// S = Qc*K^T (WMMA), softmax with running (m,l), O += P*V (WMMA).
// out[b,t, d+256j] = O[t,d] / (l * sqrt(64)),  j = 0..3  (GQA tiling).
// 8 waves: wm = wave>>1 picks 16-row strip, wn = wave&1 picks column half.
// ---------------------------------------------------------------------------
__global__ __launch_bounds__(256) void attn(const float* __restrict__ Qkv,
                                            float* __restrict__ out) {
  __shared__ float sS[64 * 68];     // padded stride 68 -> conflict-free A reads
  __shared__ float sm_[64];
  __shared__ float sl_[64];
  __shared__ float ssc[64];

  const int b   = blockIdx.y;
  const int t0  = blockIdx.x * 64;
  const int tid = threadIdx.x;
  const int lane = tid & 31;
  const int wave = tid >> 5;
  const int wm = wave >> 1;         // 0..3 : 16-row strip
  const int wn = wave & 1;          // 0..1 : column half
  const int lrow = lane & 15;
  const int lhi  = lane >> 4;

  const float* Qb = Qkv + (size_t)(b * SEQ + t0) * NW;        // Qc rows
  const float* Kb = Qkv + (size_t)(b * SEQ) * NW + 256;       // K part
  const float* Vb = Qkv + (size_t)(b * SEQ) * NW + 512;       // V part

  if (tid < 64) { sm_[tid] = -INFINITY; sl_[tid] = 0.0f; }
  __syncthreads();

  const v8f z8 = {0.f, 0.f, 0.f, 0.f, 0.f, 0.f, 0.f, 0.f};
  v8f accO[8];
#pragma unroll
  for (int f = 0; f < 8; ++f) accO[f] = z8;

  for (int s0 = 0; s0 <= t0; s0 += 64) {
    // ---- Phase A: S(64x64) = Q(64x256) * K^T(256x64), this wave: 16x32 ----
    v8f accS[2];
    accS[0] = z8; accS[1] = z8;
    for (int k0 = 0; k0 < 256; k0 += 4) {
      const int d = k0 + 2 * lhi;
      v2f a = *(const v2f*)(Qb + (size_t)(wm * 16 + lrow) * NW + d);
#pragma unroll
      for (int f = 0; f < 2; ++f) {
        const int s = s0 + wn * 32 + f * 16 + lrow;
        v2f bb = *(const v2f*)(Kb + (size_t)s * NW + d);   // K[s, d..d+1]
        accS[f] = wmma_f32(a, bb, accS[f]);
      }
    }
    // spill S to LDS (C-layout -> A-layout transpose) with causal mask
#pragma unroll
    for (int f = 0; f < 2; ++f) {
      const int col = wn * 32 + f * 16 + lrow;
      const int sg  = s0 + col;
#pragma unroll
      for (int j = 0; j < 8; ++j) {
        const int row = wm * 16 + j + 8 * lhi;
        float v = accS[f][j];
        if (sg > t0 + row) v = -INFINITY;
        sS[row * 68 + col] = v;
      }
    }
    __syncthreads();

    // ---- Phase B: per-row running softmax stats; P written back into sS ----
    if (tid < 64) {
      const int r = tid;
      float* Sr = &sS[r * 68];
      float mx = sm_[r];
      for (int c = 0; c < 64; ++c) mx = fmaxf(mx, Sr[c]);
      float sum = 0.f;
      for (int c = 0; c < 64; ++c) {
        float p = __expf(Sr[c] - mx);
        Sr[c] = p;
        sum += p;
      }
      const float sc = __expf(sm_[r] - mx);
      sm_[r]  = mx;
      sl_[r]  = sl_[r] * sc + sum;
      ssc[r]  = sc;
    }
    __syncthreads();

    // ---- Phase C: rescale O, then O(16x128) += P(16x64) * V(64x128) ----
    float scj[8];
#pragma unroll
    for (int j = 0; j < 8; ++j) scj[j] = ssc[wm * 16 + j + 8 * lhi];
#pragma unroll
    for (int f = 0; f < 8; ++f)
#pragma unroll
      for (int j = 0; j < 8; ++j) accO[f][j] *= scj[j];

    for (int k0 = 0; k0 < 64; k0 += 4) {
      const int ks = k0 + 2 * lhi;
      v2f a = *(const v2f*)(&sS[(wm * 16 + lrow) * 68 + ks]);
#pragma unroll
      for (int f = 0; f < 8; ++f) {
        const int n = wn * 128 + f * 16 + lrow;
        const float* vp = Vb + (size_t)(s0 + ks) * NW + n;
        v2f bb;
        bb.x = vp[0];
        bb.y = vp[NW];                                   // V[s+1, n]
        accO[f] = wmma_f32(a, bb, accO[f]);
      }
    }
    __syncthreads();   // protect sS before next tile overwrites it
  }

  // ---- epilogue: divide by l*sqrt(64), replicate 4x across the 1024 cols ----
  float invl[8];
#pragma unroll
  for (int j = 0; j < 8; ++j)
    invl[j] = 1.0f / (sl_[wm * 16 + j + 8 * lhi] * 8.0f);

#pragma unroll
  for (int f = 0; f < 8; ++f) {
    const int dcol = wn * 128 + f * 16 + lrow;
#pragma unroll
    for (int j = 0; j < 8; ++j) {
      const int t = t0 + wm * 16 + j + 8 * lhi;
      const float val = accO[f][j] * invl[j];
      float* op = out + (size_t)(b * SEQ + t) * 1024 + dcol;
      op[0]   = val;
      op[256] = val;
      op[512] = val;
      op[768] = val;
    }
  }
}

// ---------------------------------------------------------------------------
extern "C" void kernel_launch(void* const* d_in, const int* in_sizes, int n_in,
                              void* d_out, int out_size, void* d_ws, size_t ws_size,
                              hipStream_t stream) {
  const float* x  = (const float*)d_in[0];
  const float* Wq = (const float*)d_in[1];
  const float* bq = (const float*)d_in[2];
  const float* Wk = (const float*)d_in[3];
  const float* bk = (const float*)d_in[4];
  const float* Wv = (const float*)d_in[5];
  const float* bv = (const float*)d_in[6];
  float* out = (float*)d_out;

  // workspace layout: Wall(1024*768) | ball(768) | Qkv(16384*768)  ~51 MB
  float* Wall = (float*)d_ws;
  float* ball = Wall + (size_t)ED * NW;
  float* Qkv  = ball + NW;

  {
    const int n = ED * NW + NW;
    build_wall<<<(n + 255) / 256, 256, 0, stream>>>(Wq, bq, Wk, bk, Wv, bv, Wall, ball);
  }
  {
    dim3 grid(NW / 64, (NBATCH * SEQ) / 128);   // (12, 128)
    qkv_gemm<<<grid, 256, 0, stream>>>(x, Wall, ball, Qkv);
  }
  {
    dim3 grid(SEQ / 64, NBATCH);                // (32, 8)
    attn<<<grid, 256, 0, stream>>>(Qkv, out);
  }
}